// NucleotideSequenceEmbedding_32220844655350
// MI455X (gfx1250) — compile-verified
//
#include <hip/hip_runtime.h>
#include <math.h>

typedef __attribute__((ext_vector_type(2))) float v2f;
typedef __attribute__((ext_vector_type(8))) float v8f;

#define EMB    128
#define UNITS  64
#define GATES  256   // 4 * UNITS
#define VOCAB  5
#define LSEQ   150
#define NSEQ   4096  // B * N

// Kernel 1: one wave (32 threads). Compute h_v for all 16 padded vocab rows:
//   Z[16,256] = A(16x128) @ kernel(128x256) + bias   via V_WMMA_F32_16X16X4_F32
//   h = sigmoid(Z_o) * tanh( sigmoid(Z_i) * tanh(Z_g) )   (c0 = 0, f-gate dead)
__global__ __launch_bounds__(32)
void lstm_step0_table_kernel(const float* __restrict__ emb,     // [VOCAB, EMB]
                             const float* __restrict__ kernelW, // [EMB, GATES] row-major
                             const float* __restrict__ bias,    // [GATES]
                             float* __restrict__ h_table)       // [16, UNITS]
{
    __shared__ float Z[16 * GATES];

    const int lane = threadIdx.x;   // 0..31, EXEC all-ones (required by WMMA)
    const int m    = lane & 15;     // M (A row / D column-lane)
    const int hi   = lane >> 4;     // half-wave selector

    // Pad vocab 5 -> 16 without divergence (rows >= VOCAB contribute zero)
    const float arow_scale = (m < VOCAB) ? 1.0f : 0.0f;
    const int   arow       = (m < VOCAB) ? m : 0;

    for (int nt = 0; nt < 16; ++nt) {          // 16 column tiles of 16 (N = 256)
        v8f acc = {};
        #pragma unroll 4
        for (int k0 = 0; k0 < EMB; k0 += 4) {  // K = 128 in steps of 4
            // A 16x4 f32 layout: vgpr j, lane l -> A[l%16][k0 + j + 2*(l>>4)]
            v2f a, b;
            a.x = emb[arow * EMB + (k0 + 0 + 2 * hi)] * arow_scale;
            a.y = emb[arow * EMB + (k0 + 1 + 2 * hi)] * arow_scale;
            // B 4x16 f32 layout: vgpr j, lane l -> B[k0 + j + 2*(l>>4)][nt*16 + l%16]
            b.x = kernelW[(k0 + 0 + 2 * hi) * GATES + nt * 16 + m];
            b.y = kernelW[(k0 + 1 + 2 * hi) * GATES + nt * 16 + m];
            // 8 args: (neg_a, A, neg_b, B, c_mod, C, reuse_a, reuse_b)
            acc = __builtin_amdgcn_wmma_f32_16x16x4_f32(false, a, false, b,
                                                        (short)0, acc, false, false);
        }
        // D 16x16 f32 layout: vgpr r, lane l -> D[r + 8*(l>>4)][l%16]
        const float bv = bias[nt * 16 + m];
        #pragma unroll
        for (int r = 0; r < 8; ++r) {
            const int row = r + 8 * hi;
            Z[row * GATES + nt * 16 + m] = acc[r] + bv;
        }
    }
    __syncthreads();

    // Apply LSTM gates for step 0 (h0 = c0 = 0)
    for (int idx = lane; idx < 16 * UNITS; idx += 32) {
        const int row = idx / UNITS;
        const int u   = idx - row * UNITS;
        const float zi = Z[row * GATES + 0 * UNITS + u];
        const float zg = Z[row * GATES + 2 * UNITS + u];
        const float zo = Z[row * GATES + 3 * UNITS + u];
        const float ig = 1.0f / (1.0f + __expf(-zi));
        const float g  = tanhf(zg);
        const float og = 1.0f / (1.0f + __expf(-zo));
        const float c  = ig * g;
        h_table[idx] = og * tanhf(c);
    }
}

// Kernel 2: out[s, u] = h_table[tokens[s, 0]][u]  for s in [0,4096), u in [0,64)
__global__ __launch_bounds__(256)
void gather_h0_kernel(const int* __restrict__ tokens,   // [NSEQ, LSEQ]
                      const float* __restrict__ h_table, // [16, UNITS]
                      float* __restrict__ out)           // [NSEQ, UNITS]
{
    const int tid = blockIdx.x * blockDim.x + threadIdx.x;
    if (tid >= NSEQ * UNITS) return;
    const int s = tid / UNITS;
    const int u = tid - s * UNITS;
    const int tok = tokens[s * LSEQ];          // timestep-0 token only
    out[tid] = h_table[tok * UNITS + u];
}

extern "C" void kernel_launch(void* const* d_in, const int* in_sizes, int n_in,
                              void* d_out, int out_size, void* d_ws, size_t ws_size,
                              hipStream_t stream) {
    // setup_inputs order: tokens, emb_table, kernel, rec_kernel (dead), bias
    const int*   tokens  = (const int*)d_in[0];
    const float* emb     = (const float*)d_in[1];
    const float* kernelW = (const float*)d_in[2];
    const float* bias    = (const float*)d_in[4];
    float* out = (float*)d_out;

    float* h_table = (float*)d_ws;  // 16 * 64 * 4 = 4 KB scratch

    lstm_step0_table_kernel<<<1, 32, 0, stream>>>(emb, kernelW, bias, h_table);

    const int total = NSEQ * UNITS;
    gather_h0_kernel<<<(total + 255) / 256, 256, 0, stream>>>(tokens, h_table, out);
}